// AHFF_29317446763112
// MI455X (gfx1250) — compile-verified
//
#include <hip/hip_runtime.h>
#include <math.h>

#define B_   8
#define C_   256
#define C2   512
#define Hh   80
#define Ww   80
#define HW   6400
#define HID  32
#define K4C  1024
#define NELEM (B_*C2*HW)   // 26,214,400

typedef __attribute__((ext_vector_type(2))) float v2f;
typedef __attribute__((ext_vector_type(8))) float v8f;

__device__ __forceinline__ float fusedLoad(const float* f1, const float* f2,
                                           int b, int c, int p) {
  return (c < C_) ? f1[((size_t)b*C_ + c)*HW + p]
                  : f2[((size_t)b*C_ + (c-C_))*HW + p];
}

__device__ __forceinline__ float sigmoidf(float x) { return 1.0f/(1.0f+expf(-x)); }

// ---- 1) GAP over H,W per (b,c): one block per plane ------------------------
__global__ void gap_kernel(const float* __restrict__ f1, const float* __restrict__ f2,
                           float* __restrict__ gap) {
  int bc = blockIdx.x;                 // 0..4095
  int b = bc / C2, c = bc % C2;
  const float* src = (c < C_) ? f1 + ((size_t)b*C_+c)*HW
                              : f2 + ((size_t)b*C_ + (c-C_))*HW;
  float s = 0.f;
  for (int p = threadIdx.x; p < HW; p += 256) s += src[p];
  __shared__ float red[256];
  red[threadIdx.x] = s;
  __syncthreads();
  for (int off = 128; off > 0; off >>= 1) {
    if (threadIdx.x < off) red[threadIdx.x] += red[threadIdx.x + off];
    __syncthreads();
  }
  if (threadIdx.x == 0) gap[bc] = red[0] * (1.0f/HW);
}

// ---- 2) per-pixel mean/max over 512 channels -------------------------------
__global__ void pix_stats_kernel(const float* __restrict__ f1, const float* __restrict__ f2,
                                 float* __restrict__ sa_in) {
  int g = blockIdx.x*256 + threadIdx.x;     // 0..51199
  int b = g / HW, p = g % HW;
  const float* s1 = f1 + (size_t)b*C_*HW + p;
  const float* s2 = f2 + (size_t)b*C_*HW + p;
  float s = 0.f, m = -INFINITY;
  for (int c = 0; c < C_; ++c) { float v = s1[(size_t)c*HW]; s += v; m = fmaxf(m, v); }
  for (int c = 0; c < C_; ++c) { float v = s2[(size_t)c*HW]; s += v; m = fmaxf(m, v); }
  sa_in[((size_t)b*2+0)*HW + p] = s * (1.0f/C2);
  sa_in[((size_t)b*2+1)*HW + p] = m;
}

// ---- 3) channel-attention MLP ----------------------------------------------
__global__ void ca_mlp_kernel(const float* __restrict__ gap,
                              const float* __restrict__ w1, const float* __restrict__ b1,
                              const float* __restrict__ w2, const float* __restrict__ b2,
                              float* __restrict__ wcOut) {
  int b = blockIdx.x;
  __shared__ float sg[C2];
  __shared__ float sh[HID];
  int t = threadIdx.x;                 // 512 threads
  sg[t] = gap[(size_t)b*C2 + t];
  __syncthreads();
  if (t < HID) {
    float a = b1[t];
    for (int c = 0; c < C2; ++c) a += sg[c]*w1[(size_t)c*HID + t];
    sh[t] = fmaxf(a, 0.f);
  }
  __syncthreads();
  float a = b2[t];
  for (int j = 0; j < HID; ++j) a += sh[j]*w2[(size_t)j*C2 + t];
  wcOut[(size_t)b*C2 + t] = sigmoidf(a);
}

// ---- 4) spatial-attention 7x7 conv + sigmoid -------------------------------
__global__ void sa_conv_kernel(const float* __restrict__ sa_in,
                               const float* __restrict__ w, const float* __restrict__ bb,
                               float* __restrict__ wsm) {
  int b = blockIdx.x;
  __shared__ float sin_[2*HW];         // 51.2 KB
  __shared__ float sw[98];
  int t = threadIdx.x;                 // 256
  for (int i = t; i < 2*HW; i += 256) sin_[i] = sa_in[(size_t)b*2*HW + i];
  if (t < 98) sw[t] = w[t];
  __syncthreads();
  float bias = bb[0];
  for (int p = t; p < HW; p += 256) {
    int h = p / Ww, x = p % Ww;
    float acc = bias;
    for (int ic = 0; ic < 2; ++ic)
      for (int ky = 0; ky < 7; ++ky) {
        int hy = h + ky - 3;
        if (hy < 0 || hy >= Hh) continue;
        for (int kx = 0; kx < 7; ++kx) {
          int wx = x + kx - 3;
          if (wx < 0 || wx >= Ww) continue;
          acc += sin_[ic*HW + hy*Ww + wx] * sw[ic*49 + ky*7 + kx];
        }
      }
    wsm[(size_t)b*HW + p] = sigmoidf(acc);
  }
}

// ---- 5) fw = wc * ws * fused ----------------------------------------------
__global__ void fw_kernel(const float* __restrict__ f1, const float* __restrict__ f2,
                          const float* __restrict__ wc, const float* __restrict__ wsm,
                          float* __restrict__ fw) {
  int idx = blockIdx.x*256 + threadIdx.x;
  int b = idx / (C2*HW);
  int rem = idx - b*(C2*HW);
  int c = rem / HW, p = rem - c*HW;
  fw[idx] = wc[(size_t)b*C2+c] * wsm[(size_t)b*HW+p] * fusedLoad(f1, f2, b, c, p);
}

// ---- 6a) 80-tap circular kernel k = ifft(ifftshift(Gaussian)) --------------
__global__ void kvec_kernel(float* __restrict__ kv) {
  int t = threadIdx.x;                 // 80 threads
  const float D0 = 0.1f * 80.0f;       // ALPHA * min(H,W)
  const float inv = 1.0f/(2.0f*D0*D0);
  const float TWO_PI_OVER_80 = 0.078539816339744830962f;
  float acc = 0.f;
  for (int u = 0; u < 80; ++u) {
    float f = (u < 40) ? (float)u : (float)(u - 80);
    float g = expf(-f*f*inv);
    int ph = (u*t) % 80;               // exact phase reduction
    acc += g * cosf((float)ph * TWO_PI_OVER_80);
  }
  kv[t] = acc * (1.0f/80.0f);
}

// ---- 6b) exact HPF: hf = fw - sep. circular conv (rows then cols) ----------
__global__ void sepconv_kernel(const float* __restrict__ fw, const float* __restrict__ kv,
                               float* __restrict__ hf) {
  int bc = blockIdx.x;                 // one 80x80 plane per block
  __shared__ float sp[HW];
  __shared__ float st[HW];
  __shared__ float sk[80];
  int t = threadIdx.x;                 // 256
  const float* src = fw + (size_t)bc*HW;
  for (int i = t; i < HW/4; i += 256) {
    float4 v = reinterpret_cast<const float4*>(src)[i];
    sp[i*4+0]=v.x; sp[i*4+1]=v.y; sp[i*4+2]=v.z; sp[i*4+3]=v.w;
  }
  if (t < 80) sk[t] = kv[t];
  __syncthreads();
  for (int p = t; p < HW; p += 256) {  // row pass
    int h = p / Ww, x = p - (p/Ww)*Ww;
    float acc = 0.f;
    const float* row = sp + h*Ww;
    for (int xp = 0; xp < Ww; ++xp) {
      int d = x - xp; if (d < 0) d += Ww;
      acc += row[xp]*sk[d];
    }
    st[p] = acc;
  }
  __syncthreads();
  float* dst = hf + (size_t)bc*HW;
  for (int p = t; p < HW; p += 256) {  // col pass + subtract
    int h = p / Ww, x = p - (p/Ww)*Ww;
    float acc = 0.f;
    for (int hp = 0; hp < Hh; ++hp) {
      int d = h - hp; if (d < 0) d += Hh;
      acc += st[hp*Ww + x]*sk[d];
    }
    dst[p] = sp[p] - acc;
  }
}

// ---- 7) bias GEMM (WMMA f32) + fused epilogue out = fw + bias_b + GEMM -----
// Block tile: M=64 (o) x N=128 (p); 8 waves, each owns a 2x2 grid of 16x16
// subtiles -> 4 fragment ds-loads feed 4 WMMA per K-step.
__global__ __launch_bounds__(256) void bias_gemm_kernel(
    const float* __restrict__ fw, const float* __restrict__ hf,
    const float* __restrict__ bw, const float* __restrict__ bb,
    float* __restrict__ out) {
  const int pBase = blockIdx.x * 128;  // N tile (pixels)
  const int oBase = blockIdx.y * 64;   // M tile (out channels)
  const int b     = blockIdx.z;
  __shared__ float sA[64][33];         // [o][k], padded
  __shared__ float sB[32][132];        // [k][p], padded
  int tid  = threadIdx.x;
  int wave = tid >> 5, lane = tid & 31;
  int half = lane >> 4, l = lane & 15;
  int wm = (wave >> 2) * 32;           // 0 or 32
  int wn = (wave & 3) * 32;            // 0,32,64,96
  v8f acc00 = {0.f,0.f,0.f,0.f,0.f,0.f,0.f,0.f};
  v8f acc01 = acc00, acc10 = acc00, acc11 = acc00;
  const size_t plane = (size_t)C2*HW;
  const float* fwB = fw + (size_t)b*plane;
  const float* hfB = hf + (size_t)b*plane;

  for (int kBase = 0; kBase < K4C; kBase += 32) {
    // chunks of 32 never straddle the fw/hf boundary (512 % 32 == 0)
    const float* srcB = (kBase < C2) ? fwB + (size_t)kBase*HW
                                     : hfB + (size_t)(kBase - C2)*HW;
    // A tile 64x32 = 512 float4, 2 per thread (16B-aligned: K4C stride, kBase%32==0)
    for (int i = 0; i < 2; ++i) {
      int e = i*256 + tid;
      int r = e >> 3, c4 = (e & 7) << 2;
      float4 v = *reinterpret_cast<const float4*>(
          &bw[(size_t)(oBase + r)*K4C + kBase + c4]);
      sA[r][c4+0]=v.x; sA[r][c4+1]=v.y; sA[r][c4+2]=v.z; sA[r][c4+3]=v.w;
    }
    // B tile 32x128 = 1024 float4, 4 per thread (HW stride 25600B, pBase%128==0)
    for (int i = 0; i < 4; ++i) {
      int e = i*256 + tid;
      int r = e >> 5, c4 = (e & 31) << 2;
      float4 v = *reinterpret_cast<const float4*>(
          &srcB[(size_t)r*HW + pBase + c4]);
      sB[r][c4+0]=v.x; sB[r][c4+1]=v.y; sB[r][c4+2]=v.z; sB[r][c4+3]=v.w;
    }
    __syncthreads();
    for (int kk = 0; kk < 32; kk += 4) {
      int k0 = kk + 2*half;            // A/B frag layout: v0 K={0,2}, v1 K={1,3}
      v2f a0, a1, b0, b1;
      a0.x = sA[wm + l][k0];        a0.y = sA[wm + l][k0+1];
      a1.x = sA[wm + 16 + l][k0];   a1.y = sA[wm + 16 + l][k0+1];
      b0.x = sB[k0][wn + l];        b0.y = sB[k0+1][wn + l];
      b1.x = sB[k0][wn + 16 + l];   b1.y = sB[k0+1][wn + 16 + l];
      acc00 = __builtin_amdgcn_wmma_f32_16x16x4_f32(false, a0, false, b0,
                                                    (short)0, acc00, false, false);
      acc01 = __builtin_amdgcn_wmma_f32_16x16x4_f32(false, a0, false, b1,
                                                    (short)0, acc01, false, false);
      acc10 = __builtin_amdgcn_wmma_f32_16x16x4_f32(false, a1, false, b0,
                                                    (short)0, acc10, false, false);
      acc11 = __builtin_amdgcn_wmma_f32_16x16x4_f32(false, a1, false, b1,
                                                    (short)0, acc11, false, false);
    }
    __syncthreads();
  }
  // epilogue: out = fw + bias_b + acc   (C/D layout: VGPR r -> M = 8*half + r)
  for (int r = 0; r < 8; ++r) {
    int p0 = pBase + wn + l, p1 = p0 + 16;
    int o0 = oBase + wm + 8*half + r, o1 = o0 + 16;
    size_t base = (size_t)b*plane;
    out[base + (size_t)o0*HW + p0] = fwB[(size_t)o0*HW + p0] + bb[o0] + acc00[r];
    out[base + (size_t)o0*HW + p1] = fwB[(size_t)o0*HW + p1] + bb[o0] + acc01[r];
    out[base + (size_t)o1*HW + p0] = fwB[(size_t)o1*HW + p0] + bb[o1] + acc10[r];
    out[base + (size_t)o1*HW + p1] = fwB[(size_t)o1*HW + p1] + bb[o1] + acc11[r];
  }
}

extern "C" void kernel_launch(void* const* d_in, const int* in_sizes, int n_in,
                              void* d_out, int out_size, void* d_ws, size_t ws_size,
                              hipStream_t stream) {
  const float* feat1  = (const float*)d_in[0];
  const float* feat2  = (const float*)d_in[1];
  const float* ca_w1  = (const float*)d_in[2];
  const float* ca_b1  = (const float*)d_in[3];
  const float* ca_w2  = (const float*)d_in[4];
  const float* ca_b2  = (const float*)d_in[5];
  const float* sa_w   = (const float*)d_in[6];
  const float* sa_b   = (const float*)d_in[7];
  const float* bias_w = (const float*)d_in[8];
  const float* bias_b = (const float*)d_in[9];
  float* out = (float*)d_out;

  float* ws    = (float*)d_ws;
  float* fw    = ws;                       // NELEM
  float* hf    = fw + NELEM;               // NELEM
  float* gap   = hf + NELEM;               // B*C2
  float* wc    = gap + B_*C2;              // B*C2
  float* sa_in = wc + B_*C2;               // B*2*HW
  float* wsm   = sa_in + B_*2*HW;          // B*HW
  float* kv    = wsm + B_*HW;              // 80

  gap_kernel      <<<B_*C2,            256, 0, stream>>>(feat1, feat2, gap);
  pix_stats_kernel<<<B_*HW/256,        256, 0, stream>>>(feat1, feat2, sa_in);
  ca_mlp_kernel   <<<B_,               C2,  0, stream>>>(gap, ca_w1, ca_b1, ca_w2, ca_b2, wc);
  sa_conv_kernel  <<<B_,               256, 0, stream>>>(sa_in, sa_w, sa_b, wsm);
  fw_kernel       <<<NELEM/256,        256, 0, stream>>>(feat1, feat2, wc, wsm, fw);
  kvec_kernel     <<<1,                80,  0, stream>>>(kv);
  sepconv_kernel  <<<B_*C2,            256, 0, stream>>>(fw, kv, hf);
  bias_gemm_kernel<<<dim3(HW/128, C2/64, B_), 256, 0, stream>>>(fw, hf, bias_w, bias_b, out);
}